// LearnedEquivariance_46780783788319
// MI455X (gfx1250) — compile-verified
//
#include <hip/hip_runtime.h>

typedef __attribute__((ext_vector_type(2))) float v2f;
typedef __attribute__((ext_vector_type(8))) float v8f;

#define GROUPS 40
#define HW 56
#define PITCH 73   // 72-col window rows padded to odd pitch -> conflict-free LDS
#define WROWS 72

// One workgroup (256 threads = 8 waves) per (b, c) plane.
// The two conv passes (n=2) are fused into one 9x9 circular correlation with
// Keff = K (*) K, computed per block. Each wave computes two adjacent 16x16
// output tiles via V_WMMA_F32_16X16X4_F32. The banded-Toeplitz B matrices are
// precomputed into LDS once per block; the hot loop is pure ds_load + wmma.
__global__ __launch_bounds__(256) void le_conv9x9(
    const float* __restrict__ x, const float* __restrict__ kernels,
    const int* __restrict__ g, float* __restrict__ out, int C) {
  __shared__ float win[WROWS * PITCH];  // [72 rows x 73 pitch]
  __shared__ v2f Blds[9 * 6 * 32];      // B pairs per (di, chunk, lane)
  __shared__ float k5[25];
  __shared__ float keff[81];

  const int tid = threadIdx.x;
  const int p = blockIdx.x;  // plane = b*C + c
  const int b = p / C;
  const float* xp = x + (size_t)p * (HW * HW);
  float* op = out + (size_t)p * (HW * HW);

  // ---- Phase 1: gather this sample's 5x5 kernel ----
  if (tid < 25) {
    int gi = g[b] % GROUPS;
    if (gi < 0) gi += GROUPS;
    k5[tid] = kernels[gi * 25 + tid];
  }
  __syncthreads();

  // ---- Phase 2: compose Keff, zero apron, load wrapped 64x64 window ----
  if (tid < 81) {
    int di = tid / 9, dj = tid % 9;
    int i0 = di > 4 ? di - 4 : 0, i1 = di < 4 ? di : 4;
    int j0 = dj > 4 ? dj - 4 : 0, j1 = dj < 4 ? dj : 4;
    float s = 0.f;
    for (int i = i0; i <= i1; ++i)
      for (int j = j0; j <= j1; ++j)
        s += k5[i * 5 + j] * k5[(di - i) * 5 + (dj - j)];
    keff[tid] = s;
  }
  // Zero apron: rows 64..71 (all cols) and cols 64..72 of rows 0..63.
  // Apron values multiply exact-zero B entries; zeroing avoids 0*NaN from
  // stale LDS contaminating valid outputs.
  for (int f = tid; f < 8 * PITCH; f += 256) win[64 * PITCH + f] = 0.f;
  for (int f = tid; f < 64 * 9; f += 256) win[(f / 9) * PITCH + 64 + (f % 9)] = 0.f;
  // Main window: win[wr][wc] = xp[gr*56+gc], gr = wr-4 (+56 if wr<4), gc = wc-4 (+56 if wc<4)
  for (int f = tid; f < 64 * 64; f += 256) {
    int wr = f >> 6, wc = f & 63;
    int gr = wr + (HW - 4); if (gr >= HW) gr -= HW;
    int gc = wc + (HW - 4); if (gc >= HW) gc -= HW;
    win[wr * PITCH + wc] = xp[gr * HW + gc];
  }
  __syncthreads();

  // ---- Phase 3: precompute banded-Toeplitz B operands ----
  // B[k=xb..xb+1, n=m] for chunk ch: T[x, n] = keff[di, x - n], 0 <= x-n <= 8.
  for (int e = tid; e < 9 * 6 * 32; e += 256) {
    int di = e / 192;
    int r = e - di * 192;
    int ch = r >> 5;
    int lane = r & 31;
    int mm = lane & 15;
    int hh = lane >> 4;
    int xb = ch * 4 + hh * 2;
    int dj0 = xb - mm;
    int c0 = dj0 < 0 ? 0 : (dj0 > 8 ? 8 : dj0);
    int dj1 = dj0 + 1;
    int c1 = dj1 < 0 ? 0 : (dj1 > 8 ? 8 : dj1);
    v2f bb;
    bb.x = (dj0 == c0) ? keff[di * 9 + c0] : 0.f;
    bb.y = (dj1 == c1) ? keff[di * 9 + c1] : 0.f;
    Blds[e] = bb;
  }
  __syncthreads();

  // ---- Phase 4: WMMA compute, two 16x16 tiles per wave ----
  const int wave = tid >> 5;
  const int lane = tid & 31;
  const int m = lane & 15;     // A row (M) / D column (N)
  const int half = lane >> 4;  // K half per ISA 16x4 / 4x16 f32 layout
  const int t0 = wave * 2;     // tiles t0, t0+1 share the tile row th
  const int th = t0 >> 2, tw = t0 & 3;

  v8f acc0 = {}, acc1 = {};
  const float* ap = &win[(th * 16 + m) * PITCH + tw * 16 + half * 2];
  const v2f* bp = &Blds[lane];
  for (int di = 0; di < 9; ++di) {
#pragma unroll
    for (int ch = 0; ch < 6; ++ch) {
      v2f bb = bp[(di * 6 + ch) * 32];
      v2f a0; a0.x = ap[ch * 4]; a0.y = ap[ch * 4 + 1];
      v2f a1; a1.x = ap[16 + ch * 4]; a1.y = ap[16 + ch * 4 + 1];
      acc0 = __builtin_amdgcn_wmma_f32_16x16x4_f32(
          false, a0, false, bb, (short)0, acc0, false, false);
      acc1 = __builtin_amdgcn_wmma_f32_16x16x4_f32(
          false, a1, false, bb, (short)0, acc1, false, false);
    }
    ap += PITCH;
  }

  // Store per documented D layout: lane -> N = lane&15, VGPR v -> M = v + 8*half
  const int oc0 = tw * 16 + m;
  const int oc1 = oc0 + 16;
#pragma unroll
  for (int v = 0; v < 8; ++v) {
    int orr = th * 16 + v + half * 8;
    if (orr < HW) {
      if (oc0 < HW) op[orr * HW + oc0] = acc0[v];
      if (oc1 < HW) op[orr * HW + oc1] = acc1[v];
    }
  }
}

extern "C" void kernel_launch(void* const* d_in, const int* in_sizes, int n_in,
                              void* d_out, int out_size, void* d_ws, size_t ws_size,
                              hipStream_t stream) {
  const float* x = (const float*)d_in[0];        // [B, C, 56, 56] fp32
  const float* kernels = (const float*)d_in[1];  // [40, 5, 5] fp32
  const int* g = (const int*)d_in[2];            // [B] int32 group ids
  // d_in[3] = l (0), d_in[4] = n (2): static; n=2 fused into composed kernel.
  float* out = (float*)d_out;

  int planes = in_sizes[0] / (HW * HW);  // B*C = 8192
  int B = in_sizes[2];                   // 128
  int C = planes / B;                    // 64

  le_conv9x9<<<planes, 256, 0, stream>>>(x, kernels, g, out, C);
}